// ChamferAugmented_16346645528639
// MI455X (gfx1250) — compile-verified
//
#include <hip/hip_runtime.h>
#include <float.h>

// ChamferAugmented for MI455X (gfx1250, wave32).
// dist^2(g,p) = |g|^2 + ( |p|^2 - 2 g.p ) -> K=4 dot product per pair:
//   A[p]   = (px, py, pz, |p|^2)       (16x4 f32, points on M axis)
//   B[:,g] = (-2gx, -2gy, -2gz, 1)     (4x16 f32, grid samples on N axis)
// V_WMMA_F32_16X16X4_F32 -> 16x16 tile of S[p][g]; min over p via v_min3_num.
// Points are pre-packed once into A-operand-ready 16B records so the hot loop
// is 4 WMMA + 16 min3 + 4 b64 loads (immediate offsets only).
// 8 point-chunks per (batch, tile) -> 8192 waves (~8/SIMD32) to hide L2 load
// latency; partial mins merge via GLOBAL_ATOMIC_MIN_NUM_F32 (exact,
// order-independent => deterministic).

typedef float v2f __attribute__((ext_vector_type(2)));
typedef float v8f __attribute__((ext_vector_type(8)));

#define BS        8
#define N_PTS     8192
#define N_SAMPLES 2048
#define N_CHUNKS  8
#define TILES_PER_CHUNK (N_PTS / 16 / N_CHUNKS)   // 64 point-tiles per chunk
#define PTS_PER_SET     (BS * N_PTS)              // 65536

__global__ __launch_bounds__(256) void chamfer_init_kernel(float* __restrict__ mins) {
    const int idx = blockIdx.x * 256 + threadIdx.x;   // 2 * BS * N_SAMPLES floats
    mins[idx] = FLT_MAX;
}

// Pack each point once as (x, y, z, |p|^2): set 0 = preds, set 1 = gts.
__global__ __launch_bounds__(256) void chamfer_prep_kernel(
    const float* __restrict__ gts, const float* __restrict__ preds,
    float4* __restrict__ aug)
{
    const int idx = blockIdx.x * 256 + threadIdx.x;   // 0 .. 2*PTS_PER_SET-1
    const int set = idx >> 16;                        // /65536
    const int rem = idx & (PTS_PER_SET - 1);
    const float* src = (set ? gts : preds) + (size_t)rem * 3;
    const float x = src[0], y = src[1], z = src[2];
    aug[idx] = make_float4(x, y, z, x * x + y * y + z * z);
}

__global__ __launch_bounds__(256) void chamfer_min_kernel(
    const float* __restrict__ aug,    // (2, BS, N_PTS, 4) packed A operands
    const float* __restrict__ grid,   // (BS, N_SAMPLES, 3)
    float* __restrict__ minP,         // (BS, N_SAMPLES) init +INF
    float* __restrict__ minG)         // (BS, N_SAMPLES) init +INF
{
    const int lane = threadIdx.x & 31;
    const int wave = threadIdx.x >> 5;
    const int widx = blockIdx.x * 8 + wave;     // 0..8191
    const int b     = widx >> 10;               // batch 0..7
    const int rem   = widx & 1023;
    const int tile  = rem >> 3;                 // grid tile 0..127
    const int chunk = rem & 7;                  // point chunk 0..7

    const int sub = lane & 15;
    const int hi  = lane >> 4;                  // 0 -> K=0,1 half; 1 -> K=2,3 half
    const int s   = tile * 16 + sub;            // sample index

    // ---- B operand (grid tile): fixed for the whole wave ----
    const float* gp = grid + ((size_t)b * N_SAMPLES + s) * 3;
    const float gx = gp[0], gy = gp[1], gz = gp[2];
    v2f bmat;
    bmat.x = hi ? (-2.0f * gz) : (-2.0f * gx);  // K=2 : K=0
    bmat.y = hi ? 1.0f         : (-2.0f * gy);  // K=3 : K=1

    // Loop-invariant per-lane A-operand base: lane reads floats (0,1) or (2,3)
    // of its point's 16B record -> one global_load_b64 per tile, imm offsets.
    const size_t pbase = (size_t)b * N_PTS + (size_t)chunk * (N_PTS / N_CHUNKS);
    const float* AP = aug + (pbase + sub) * 4 + hi * 2;          // preds
    const float* AG = AP + (size_t)PTS_PER_SET * 4;              // gts

    float mp[8], mg[8];
#pragma unroll
    for (int i = 0; i < 8; ++i) { mp[i] = FLT_MAX; mg[i] = FLT_MAX; }

    v8f czero = {};
    // Two point-tiles per step: 4 WMMAs + 16 v_min3 + 4 b64 loads.
    for (int t = 0; t < TILES_PER_CHUNK; t += 2) {
        const v2f a0 = *(const v2f*)(AP + (size_t)t * 64);
        const v2f a1 = *(const v2f*)(AP + (size_t)t * 64 + 64);
        const v2f c0 = *(const v2f*)(AG + (size_t)t * 64);
        const v2f c1 = *(const v2f*)(AG + (size_t)t * 64 + 64);

        v8f d0 = __builtin_amdgcn_wmma_f32_16x16x4_f32(
            false, a0, false, bmat, (short)0, czero, false, false);
        v8f d1 = __builtin_amdgcn_wmma_f32_16x16x4_f32(
            false, a1, false, bmat, (short)0, czero, false, false);
        v8f e0 = __builtin_amdgcn_wmma_f32_16x16x4_f32(
            false, c0, false, bmat, (short)0, czero, false, false);
        v8f e1 = __builtin_amdgcn_wmma_f32_16x16x4_f32(
            false, c1, false, bmat, (short)0, czero, false, false);

#pragma unroll
        for (int i = 0; i < 8; ++i) {
            asm("v_min3_num_f32 %0, %0, %1, %2" : "+v"(mp[i]) : "v"(d0[i]), "v"(d1[i]));
            asm("v_min3_num_f32 %0, %0, %1, %2" : "+v"(mg[i]) : "v"(e0[i]), "v"(e1[i]));
        }
    }

    // Reduce over M (points): 8 regs in-lane, then across wave halves.
    float rp = mp[0], rg = mg[0];
#pragma unroll
    for (int i = 1; i < 8; i += 2) {
        asm("v_min3_num_f32 %0, %0, %1, %2" : "+v"(rp) : "v"(mp[i]), "v"(mp[i + 1]));
        asm("v_min3_num_f32 %0, %0, %1, %2" : "+v"(rg) : "v"(mg[i]), "v"(mg[i + 1]));
    }
    rp = fminf(rp, __shfl_xor(rp, 16, 32));
    rg = fminf(rg, __shfl_xor(rg, 16, 32));

    // Merge partial (|p|^2 - 2 g.p) mins across the 8 point chunks.
    if (hi == 0) {
        float* ap = minP + (size_t)b * N_SAMPLES + s;
        float* ag = minG + (size_t)b * N_SAMPLES + s;
        __hip_atomic_fetch_min(ap, rp, __ATOMIC_RELAXED, __HIP_MEMORY_SCOPE_AGENT);
        __hip_atomic_fetch_min(ag, rg, __ATOMIC_RELAXED, __HIP_MEMORY_SCOPE_AGENT);
    }
}

__global__ __launch_bounds__(256) void chamfer_reduce_kernel(
    const float* __restrict__ minP, const float* __restrict__ minG,
    const float* __restrict__ grid, float* __restrict__ out)
{
    __shared__ float sm[256];
    const int b = blockIdx.x;
    float acc = 0.0f;
    for (int s = threadIdx.x; s < N_SAMPLES; s += 256) {
        const float* gp = grid + ((size_t)b * N_SAMPLES + s) * 3;
        const float gx = gp[0], gy = gp[1], gz = gp[2];
        const float g2 = gx * gx + gy * gy + gz * gz;
        const float dp = sqrtf(fmaxf(minP[(size_t)b * N_SAMPLES + s] + g2, 0.0f));
        const float dg = sqrtf(fmaxf(minG[(size_t)b * N_SAMPLES + s] + g2, 0.0f));
        acc += fabsf(dp - dg);
    }
    sm[threadIdx.x] = acc;
    __syncthreads();
    for (int stride = 128; stride > 0; stride >>= 1) {
        if (threadIdx.x < stride) sm[threadIdx.x] += sm[threadIdx.x + stride];
        __syncthreads();
    }
    if (threadIdx.x == 0) out[b] = sm[0] * (1.0f / (float)N_SAMPLES);
}

extern "C" void kernel_launch(void* const* d_in, const int* in_sizes, int n_in,
                              void* d_out, int out_size, void* d_ws, size_t ws_size,
                              hipStream_t stream) {
    const float* gts   = (const float*)d_in[0];  // (8, 8192, 3)
    const float* preds = (const float*)d_in[1];  // (8, 8192, 3)
    const float* grid  = (const float*)d_in[2];  // (8, 2048, 3)
    float*       out   = (float*)d_out;          // (8,)

    float* minP = (float*)d_ws;                            // (8, 2048)  64 KB
    float* minG = minP + (size_t)BS * N_SAMPLES;           // (8, 2048)  64 KB
    float* aug  = minG + (size_t)BS * N_SAMPLES;           // (2,8,8192,4) 2 MB

    // Re-init min buffers to +INF every call (atomic mins only ever decrease).
    chamfer_init_kernel<<<dim3(2 * BS * N_SAMPLES / 256), dim3(256), 0, stream>>>(minP);

    // Pack points once into WMMA-A-ready 16B records.
    chamfer_prep_kernel<<<dim3(2 * PTS_PER_SET / 256), dim3(256), 0, stream>>>(
        gts, preds, (float4*)aug);

    // 8192 waves = 8 batches x 128 grid tiles x 8 point chunks; 8 waves/block.
    chamfer_min_kernel<<<dim3(1024), dim3(256), 0, stream>>>(aug, grid, minP, minG);

    // Deterministic per-batch |distP - distG| mean.
    chamfer_reduce_kernel<<<dim3(BS), dim3(256), 0, stream>>>(minP, minG, grid, out);
}